// Backbone_4449586118721
// MI455X (gfx1250) — compile-verified
//
#include <hip/hip_runtime.h>
#include <math.h>

// ---------------------------------------------------------------------------
// Backbone for CDNA5 / gfx1250 (wave32). All GEMMs use V_WMMA_F32_16X16X4_F32
// (full f32 precision on the matrix pipe). Patch-embed path is algebraically
// folded so the huge z_p intermediate is never materialized. Generic GEMM
// computes NT=4 (or 2) column tiles per wave so each A fragment feeds NT
// independent WMMA chains (4x less A traffic, better latency hiding).
// ---------------------------------------------------------------------------

#define B_    256
#define L_    512
#define D_    21
#define BD_   (B_ * D_)      // 5376
#define N_    32
#define P_    16
#define PRED_ 96
#define EPS_BN 1e-5f

typedef __attribute__((ext_vector_type(2))) float v2f;
typedef __attribute__((ext_vector_type(8))) float v8f;

__device__ __forceinline__ float gelu_exact(float x) {
  return 0.5f * x * (1.0f + erff(x * 0.70710678118654752f));
}

// ---------------------------------------------------------------------------
// Generic row-major f32 GEMM: C = act(A(MxK) @ B(KxN) + bias + resid)
// One wave computes a 16 x (16*NT) tile via K/4 chained v_wmma_f32_16x16x4_f32.
// Fragment layouts per CDNA5 ISA 7.12.2:
//   A 16x4 : lane&15 = M row; VGPR0/1 hold K = kk, kk+1 where kk = 2*(lane>>4)
//   B 4x16 : lane&15 = N col; VGPR0/1 hold K rows kk, kk+1
//   C 16x16: VGPR v -> M = v + 8*(lane>>4), N = lane&15
// ---------------------------------------------------------------------------
template <int NT>
__global__ void k_gemm_wmma(const float* __restrict__ A, const float* __restrict__ Bm,
                            float* __restrict__ C, const float* __restrict__ bias,
                            const float* __restrict__ resid,
                            int M, int Kd, int Nc, int act) {
  int wave = (int)((blockIdx.x * blockDim.x + threadIdx.x) >> 5);
  int lane = (int)(threadIdx.x & 31);
  int tilesN = Nc >> 4;
  int gpr = tilesN / NT;                     // tile-groups along N
  int groups = (M >> 4) * gpr;
  if (wave >= groups) return;                // wave-uniform: EXEC stays all-ones
  int tm = wave / gpr;
  int tn0 = (wave % gpr) * NT;
  const float* Arow = A + (size_t)(tm * 16 + (lane & 15)) * Kd;
  int kk = (lane >> 4) * 2;
  int ncol = lane & 15;
  v8f acc[NT];
#pragma unroll
  for (int t = 0; t < NT; ++t) { v8f z = {}; acc[t] = z; }
  for (int k0 = 0; k0 < Kd; k0 += 4) {
    v2f a;
    a.x = Arow[k0 + kk];
    a.y = Arow[k0 + kk + 1];
    const float* Brow0 = Bm + (size_t)(k0 + kk) * Nc + tn0 * 16 + ncol;
    const float* Brow1 = Bm + (size_t)(k0 + kk + 1) * Nc + tn0 * 16 + ncol;
#pragma unroll
    for (int t = 0; t < NT; ++t) {
      v2f b;
      b.x = Brow0[t * 16];
      b.y = Brow1[t * 16];
      acc[t] = __builtin_amdgcn_wmma_f32_16x16x4_f32(false, a, false, b, (short)0,
                                                     acc[t], false, false);
    }
  }
  int rowBase = tm * 16 + ((lane >> 4) ? 8 : 0);
#pragma unroll
  for (int t = 0; t < NT; ++t) {
    int col = (tn0 + t) * 16 + ncol;
    float bv = bias ? bias[col] : 0.0f;
    for (int v = 0; v < 8; ++v) {
      int r = rowBase + v;
      float val = acc[t][v] + bv;
      if (resid) val += resid[(size_t)r * Nc + col];
      if (act) val = gelu_exact(val);
      C[(size_t)r * Nc + col] = val;
    }
  }
}

// Block-diagonal patch GEMM: S[r, n*16+q] = xT[r, n*16 : n*16+16] @ Ws[n] + bs[n]
__global__ void k_gemm_patch(const float* __restrict__ xT, const float* __restrict__ Ws,
                             const float* __restrict__ bs, float* __restrict__ S) {
  int wave = (int)((blockIdx.x * blockDim.x + threadIdx.x) >> 5);
  int lane = (int)(threadIdx.x & 31);
  if (wave >= (BD_ / 16) * N_) return;
  int tm = wave >> 5;          // 0..335
  int n  = wave & 31;          // 0..31
  const float* Arow = xT + (size_t)(tm * 16 + (lane & 15)) * L_ + n * 16;
  const float* Bn = Ws + n * 256;            // 16x16 row-major [t][q]
  int kk = (lane >> 4) * 2;
  v8f acc = {};
  for (int k0 = 0; k0 < 16; k0 += 4) {
    v2f a, b;
    a.x = Arow[k0 + kk];
    a.y = Arow[k0 + kk + 1];
    b.x = Bn[(k0 + kk) * 16 + (lane & 15)];
    b.y = Bn[(k0 + kk + 1) * 16 + (lane & 15)];
    acc = __builtin_amdgcn_wmma_f32_16x16x4_f32(false, a, false, b, (short)0, acc,
                                                false, false);
  }
  int q = lane & 15;
  float bv = bs[n * 16 + q];
  int rowBase = tm * 16 + ((lane >> 4) ? 8 : 0);
  for (int v = 0; v < 8; ++v)
    S[(size_t)(rowBase + v) * L_ + n * 16 + q] = acc[v] + bv;
}

// ---------------------------------------------------------------------------
// DCT matrix generation (replicates reference f32 op order exactly)
// B1[m][l] = 2*cos(pi*(2m+1)*l/1024)                (DCT-II, B of xT @ D2^T)
// B2[k][m] = cos(pi*(2m+1)*k/1024)*w0(k)/512        (DCT-III, B of X @ D3^T)
// ---------------------------------------------------------------------------
__global__ void k_gen_dct(float* B1, float* B2) {
  int i = blockIdx.x * blockDim.x + threadIdx.x;
  if (i >= L_ * L_) return;
  int r = i >> 9, c = i & 511;
  const float pf = 3.14159265358979323846f;
  B1[i] = 2.0f * cosf(((pf * (2.0f * r + 1.0f)) * (float)c) * (1.0f / 1024.0f));
  float w0 = (r == 0) ? 0.5f : 1.0f;
  B2[i] = cosf(((pf * (2.0f * c + 1.0f)) * (float)r) * (1.0f / 1024.0f)) * w0 *
          (1.0f / 512.0f);
}

__global__ void k_transpose(const float* __restrict__ x, const float* __restrict__ w_dct,
                            const float* __restrict__ b_dct, float* __restrict__ xT,
                            float* __restrict__ xres) {
  int i = blockIdx.x * blockDim.x + threadIdx.x;
  if (i >= BD_ * L_) return;
  int r = i / L_, l = i % L_;
  int b = r / D_, d = r % D_;
  float v = x[(size_t)(b * L_ + l) * D_ + d];
  xT[i] = v;
  xres[i] = v * w_dct[d] + b_dct[d];
}

__global__ void k_rowsumsq(const float* __restrict__ Z, float* __restrict__ energy) {
  int wave = (int)((blockIdx.x * blockDim.x + threadIdx.x) >> 5);
  int lane = (int)(threadIdx.x & 31);
  if (wave >= BD_) return;
  const float* row = Z + (size_t)wave * L_;
  float s = 0.f;
  for (int l = lane; l < L_; l += 32) { float v = row[l]; s += v * v; }
  for (int off = 16; off > 0; off >>= 1) s += __shfl_down(s, off, 32);
  if (lane == 0) energy[wave] = s;
}

__global__ void k_median(const float* __restrict__ energy, float* __restrict__ norme) {
  int b = blockIdx.x * blockDim.x + threadIdx.x;
  if (b >= B_) return;
  float a[D_], srt[D_];
  for (int d = 0; d < D_; ++d) { a[d] = energy[b * D_ + d]; srt[d] = a[d]; }
  for (int i = 1; i < D_; ++i) {
    float key = srt[i]; int j = i - 1;
    while (j >= 0 && srt[j] > key) { srt[j + 1] = srt[j]; --j; }
    srt[j + 1] = key;
  }
  float inv = 1.0f / (srt[D_ / 2] + 1e-6f);   // D_=21 odd -> element 10
  for (int d = 0; d < D_; ++d) norme[b * D_ + d] = a[d] * inv;
}

// Single-workgroup bitonic sort of 5376 values (padded to 8192) -> linear quantile
__global__ void k_quantile(const float* __restrict__ norme, const float* __restrict__ thr_p,
                           float* __restrict__ scal) {
  __shared__ float arr[8192];
  int tid = threadIdx.x;   // 1024 threads
  for (int i = tid; i < 8192; i += 1024) arr[i] = (i < BD_) ? norme[i] : 3.4e38f;
  __syncthreads();
  for (int k = 2; k <= 8192; k <<= 1) {
    for (int j = k >> 1; j > 0; j >>= 1) {
      for (int i = tid; i < 8192; i += 1024) {
        int ixj = i ^ j;
        if (ixj > i) {
          float a = arr[i], b = arr[ixj];
          bool up = ((i & k) == 0);
          if (up ? (a > b) : (a < b)) { arr[i] = b; arr[ixj] = a; }
        }
      }
      __syncthreads();
    }
  }
  if (tid == 0) {
    float pos = thr_p[0] * (float)(BD_ - 1);
    int lo = (int)floorf(pos);
    if (lo < 0) lo = 0;
    if (lo > BD_ - 2) lo = BD_ - 2;
    float frac = pos - (float)lo;
    scal[0] = arr[lo] + frac * (arr[lo + 1] - arr[lo]);
  }
}

__global__ void k_mask_gelu(float* __restrict__ Z, const float* __restrict__ norme,
                            const float* __restrict__ scal, const float* __restrict__ w_dct,
                            const float* __restrict__ b_dct) {
  int i = blockIdx.x * blockDim.x + threadIdx.x;
  if (i >= BD_ * L_) return;
  int r = i / L_;
  float m = (norme[r] > scal[0]) ? 1.0f : 0.0f;
  int d = r % D_;
  Z[i] = gelu_exact(Z[i] * m * w_dct[d] + b_dct[d]);
}

// BN stats, channel = (row % 21), reduce over (B, L)
__global__ void k_bn_stats_dct(const float* __restrict__ Z, float* __restrict__ st) {
  int d = blockIdx.x, tid = threadIdx.x;
  float s = 0.f, s2 = 0.f;
  for (int b = 0; b < B_; ++b) {
    const float* row = Z + (size_t)(b * D_ + d) * L_;
    for (int l = tid; l < L_; l += 256) { float v = row[l]; s += v; s2 += v * v; }
  }
  __shared__ float sh[256], sh2[256];
  sh[tid] = s; sh2[tid] = s2; __syncthreads();
  for (int o = 128; o > 0; o >>= 1) {
    if (tid < o) { sh[tid] += sh[tid + o]; sh2[tid] += sh2[tid + o]; }
    __syncthreads();
  }
  if (tid == 0) {
    float n = (float)(B_ * L_);
    float mean = sh[0] / n;
    st[d * 2] = mean;
    st[d * 2 + 1] = sh2[0] / n - mean * mean;   // biased var (jnp default)
  }
}

__global__ void k_bn_apply_dct(float* __restrict__ Z, const float* __restrict__ st,
                               const float* __restrict__ g, const float* __restrict__ bb) {
  int i = blockIdx.x * blockDim.x + threadIdx.x;
  if (i >= BD_ * L_) return;
  int d = (i / L_) % D_;
  Z[i] = g[d] * (Z[i] - st[d * 2]) * rsqrtf(st[d * 2 + 1] + EPS_BN) + bb[d];
}

// ------------------- folded patch-path weight preparation -------------------
// W_eff[n*16+p][j] = sum_q W_embed[p][q] * W_linres[n*256+q][j]
__global__ void k_prep_weff(const float* __restrict__ W_embed,
                            const float* __restrict__ W_linres, float* __restrict__ Weff) {
  int i = blockIdx.x * blockDim.x + threadIdx.x;
  if (i >= 512 * PRED_) return;
  int row = i / PRED_, j = i % PRED_;
  int n = row >> 4, p = row & 15;
  float s = 0.f;
  for (int q = 0; q < 256; ++q)
    s += W_embed[p * 256 + q] * W_linres[(size_t)(n * 256 + q) * PRED_ + j];
  Weff[i] = s;
}

__global__ void k_prep_beff(const float* __restrict__ b_embed,
                            const float* __restrict__ W_linres,
                            const float* __restrict__ b_linres, float* __restrict__ beff) {
  int j = blockIdx.x * blockDim.x + threadIdx.x;
  if (j >= PRED_) return;
  float s = b_linres[j];
  for (int n = 0; n < N_; ++n)
    for (int q = 0; q < 256; ++q)
      s += b_embed[q] * W_linres[(size_t)(n * 256 + q) * PRED_ + j];
  beff[j] = s;
}

// Ws[n][t][q] = sum_p W_embed[t][q*16+p]*w_dc[n][p] + sum_qq W_embed[t][qq]*W_dres[qq][q]
__global__ void k_prep_ws(const float* __restrict__ W_embed, const float* __restrict__ w_dc,
                          const float* __restrict__ W_dres, float* __restrict__ Ws) {
  int i = blockIdx.x * blockDim.x + threadIdx.x;
  if (i >= N_ * 256) return;
  int n = i >> 8, t = (i >> 4) & 15, q = i & 15;
  float wd = 0.f;
  for (int p = 0; p < 16; ++p) wd += W_embed[t * 256 + q * 16 + p] * w_dc[n * 16 + p];
  float wer = 0.f;
  for (int qq = 0; qq < 256; ++qq) wer += W_embed[t * 256 + qq] * W_dres[qq * 16 + q];
  Ws[i] = wd + wer;
}

__global__ void k_prep_bs(const float* __restrict__ b_embed, const float* __restrict__ w_dc,
                          const float* __restrict__ b_dc, const float* __restrict__ W_dres,
                          const float* __restrict__ b_dres, float* __restrict__ bs) {
  int i = blockIdx.x * blockDim.x + threadIdx.x;
  if (i >= N_ * 16) return;
  int n = i >> 4, q = i & 15;
  float s = b_dc[n] + b_dres[q];
  for (int p = 0; p < 16; ++p) s += b_embed[q * 16 + p] * w_dc[n * 16 + p];
  for (int qq = 0; qq < 256; ++qq) s += b_embed[qq] * W_dres[qq * 16 + q];
  bs[i] = s;
}

// BN stats, channel = (col / 16), reduce over (rows, 16 cols)
__global__ void k_bn_stats_cols(const float* __restrict__ Z, float* __restrict__ st) {
  int n = blockIdx.x, tid = threadIdx.x;
  float s = 0.f, s2 = 0.f;
  for (int r = tid; r < BD_; r += 256) {
    const float* p = Z + (size_t)r * L_ + n * 16;
    for (int q = 0; q < 16; ++q) { float v = p[q]; s += v; s2 += v * v; }
  }
  __shared__ float sh[256], sh2[256];
  sh[tid] = s; sh2[tid] = s2; __syncthreads();
  for (int o = 128; o > 0; o >>= 1) {
    if (tid < o) { sh[tid] += sh[tid + o]; sh2[tid] += sh2[tid + o]; }
    __syncthreads();
  }
  if (tid == 0) {
    float ne = (float)(BD_ * 16);
    float mean = sh[0] / ne;
    st[n * 2] = mean;
    st[n * 2 + 1] = sh2[0] / ne - mean * mean;
  }
}

__global__ void k_bn_gelu_cols(float* __restrict__ Z, const float* __restrict__ st,
                               const float* __restrict__ g, const float* __restrict__ bb) {
  int i = blockIdx.x * blockDim.x + threadIdx.x;
  if (i >= BD_ * L_) return;
  int n = (i % L_) >> 4;
  Z[i] = gelu_exact(g[n] * (Z[i] - st[n * 2]) * rsqrtf(st[n * 2 + 1] + EPS_BN) + bb[n]);
}

__global__ void k_dconv(const float* __restrict__ Z2, const float* __restrict__ w_dc1,
                        const float* __restrict__ b_dc1, float* __restrict__ ZC) {
  int i = blockIdx.x * blockDim.x + threadIdx.x;
  if (i >= BD_ * L_) return;
  int c = i % L_, n = c >> 4, ii = c & 15;
  float left  = (ii > 0)  ? Z2[i - 1] : 0.f;
  float right = (ii < 15) ? Z2[i + 1] : 0.f;
  ZC[i] = w_dc1[n * 3] * left + w_dc1[n * 3 + 1] * Z2[i] + w_dc1[n * 3 + 2] * right +
          b_dc1[n];
}

__global__ void k_bn_gelu_add_cols(float* __restrict__ ZC, const float* __restrict__ Z2,
                                   const float* __restrict__ st, const float* __restrict__ g,
                                   const float* __restrict__ bb) {
  int i = blockIdx.x * blockDim.x + threadIdx.x;
  if (i >= BD_ * L_) return;
  int n = (i % L_) >> 4;
  float v = g[n] * (ZC[i] - st[n * 2]) * rsqrtf(st[n * 2 + 1] + EPS_BN) + bb[n];
  ZC[i] = gelu_exact(v) + Z2[i];
}

// att[r] = sum_k (sum_m z1[r,m]*z2f[r,m]*B1[m][k]) * s_ortho[k] * a_dw_w[k] + a_dw_b
__global__ void k_att(const float* __restrict__ Z1, const float* __restrict__ ZC,
                      const float* __restrict__ B1, const float* __restrict__ a_dw_w,
                      const float* __restrict__ a_dw_b, float* __restrict__ att) {
  int wave = (int)((blockIdx.x * blockDim.x + threadIdx.x) >> 5);
  int lane = (int)(threadIdx.x & 31);
  if (wave >= BD_) return;
  const float* r1 = Z1 + (size_t)wave * L_;
  const float* r2 = ZC + (size_t)wave * L_;
  float a0 = 0, a1 = 0, a2 = 0, a3 = 0, a4 = 0;
  for (int l = lane; l < L_; l += 32) {
    float p = r1[l] * r2[l];
    const float* bm = B1 + (size_t)l * L_;
    a0 += p * bm[0]; a1 += p * bm[1]; a2 += p * bm[2]; a3 += p * bm[3]; a4 += p * bm[4];
  }
  for (int off = 16; off > 0; off >>= 1) {
    a0 += __shfl_down(a0, off, 32);
    a1 += __shfl_down(a1, off, 32);
    a2 += __shfl_down(a2, off, 32);
    a3 += __shfl_down(a3, off, 32);
    a4 += __shfl_down(a4, off, 32);
  }
  if (lane == 0) {
    const float rL = 22.62741699796952f;               // sqrt(512)
    const float s0 = 0.5f / rL, sk = 0.70710678118654752f / rL;
    att[wave] = a0 * s0 * a_dw_w[0] +
                (a1 * a_dw_w[1] + a2 * a_dw_w[2] + a3 * a_dw_w[3] + a4 * a_dw_w[4]) * sk +
                a_dw_b[0];
  }
}

__global__ void k_att_stats(const float* __restrict__ att, float* __restrict__ scal) {
  int tid = threadIdx.x;   // 1024
  float s = 0.f, s2 = 0.f;
  for (int i = tid; i < BD_; i += 1024) { float v = att[i]; s += v; s2 += v * v; }
  __shared__ float sh[1024], sh2[1024];
  sh[tid] = s; sh2[tid] = s2; __syncthreads();
  for (int o = 512; o > 0; o >>= 1) {
    if (tid < o) { sh[tid] += sh[tid + o]; sh2[tid] += sh2[tid + o]; }
    __syncthreads();
  }
  if (tid == 0) {
    float n = (float)BD_;
    float mean = sh[0] / n;
    scal[1] = mean;
    scal[2] = sh2[0] / n - mean * mean;
  }
}

__global__ void k_att_softmax(const float* __restrict__ att, const float* __restrict__ scal,
                              const float* __restrict__ a_g, const float* __restrict__ a_b,
                              const float* __restrict__ a_cw, const float* __restrict__ a_cb,
                              float* __restrict__ att1) {
  int b = blockIdx.x * blockDim.x + threadIdx.x;
  if (b >= B_) return;
  float m = scal[1], rs = rsqrtf(scal[2] + EPS_BN);
  float lg[D_], mx = -3.4e38f;
  for (int d = 0; d < D_; ++d) {
    float t = a_g[0] * (att[b * D_ + d] - m) * rs + a_b[0];
    t = gelu_exact(t);
    lg[d] = t * a_cw[0] + a_cb[0];
    mx = fmaxf(mx, lg[d]);
  }
  float sum = 0.f;
  for (int d = 0; d < D_; ++d) { lg[d] = expf(lg[d] - mx); sum += lg[d]; }
  float inv = 1.0f / sum;
  for (int d = 0; d < D_; ++d) att1[b * D_ + d] = lg[d] * inv;
}

__global__ void k_zsum(const float* __restrict__ Z1, const float* __restrict__ ZC,
                       const float* __restrict__ att1, float* __restrict__ Zout) {
  int i = blockIdx.x * blockDim.x + threadIdx.x;
  if (i >= BD_ * L_) return;
  float a = att1[i / L_];
  Zout[i] = Z1[i] * a + ZC[i] * (1.0f - a);
}

// out[b, j, d] = z_res[(b*21+d), j] + z_mlp[(b*21+d), j]
__global__ void k_final(const float* __restrict__ zres, const float* __restrict__ zmlp,
                        float* __restrict__ out) {
  int i = blockIdx.x * blockDim.x + threadIdx.x;
  if (i >= B_ * PRED_ * D_) return;
  int d = i % D_, j = (i / D_) % PRED_, b = i / (D_ * PRED_);
  size_t src = (size_t)(b * D_ + d) * PRED_ + j;
  out[i] = zres[src] + zmlp[src];
}

// ---------------------------------------------------------------------------
#define CDIV(a, b) (((a) + (b) - 1) / (b))

static void launch_gemm(const float* A, const float* Bm, float* C, const float* bias,
                        const float* resid, int M, int Kd, int Nc, int act,
                        hipStream_t stream) {
  int tilesN = Nc / 16;
  if (tilesN % 4 == 0) {
    int groups = (M / 16) * (tilesN / 4);
    k_gemm_wmma<4><<<CDIV(groups, 8), 256, 0, stream>>>(A, Bm, C, bias, resid, M, Kd, Nc, act);
  } else if (tilesN % 2 == 0) {
    int groups = (M / 16) * (tilesN / 2);
    k_gemm_wmma<2><<<CDIV(groups, 8), 256, 0, stream>>>(A, Bm, C, bias, resid, M, Kd, Nc, act);
  } else {
    int groups = (M / 16) * tilesN;
    k_gemm_wmma<1><<<CDIV(groups, 8), 256, 0, stream>>>(A, Bm, C, bias, resid, M, Kd, Nc, act);
  }
}

extern "C" void kernel_launch(void* const* d_in, const int* in_sizes, int n_in,
                              void* d_out, int out_size, void* d_ws, size_t ws_size,
                              hipStream_t stream) {
  (void)in_sizes; (void)n_in; (void)out_size; (void)ws_size;
  const float* x        = (const float*)d_in[0];
  const float* w_dct    = (const float*)d_in[1];
  const float* b_dct    = (const float*)d_in[2];
  const float* W_embed  = (const float*)d_in[3];
  const float* b_embed  = (const float*)d_in[4];
  const float* W_linres = (const float*)d_in[5];
  const float* b_linres = (const float*)d_in[6];
  const float* W_dres   = (const float*)d_in[7];
  const float* b_dres   = (const float*)d_in[8];
  const float* w_dc     = (const float*)d_in[9];
  const float* b_dc     = (const float*)d_in[10];
  const float* w_dc1    = (const float*)d_in[11];
  const float* b_dc1    = (const float*)d_in[12];
  const float* g_dctn   = (const float*)d_in[13];
  const float* b_dctn   = (const float*)d_in[14];
  const float* g_patch  = (const float*)d_in[15];
  const float* b_patch  = (const float*)d_in[16];
  const float* g_depth  = (const float*)d_in[17];
  const float* b_depth  = (const float*)d_in[18];
  const float* thr_p    = (const float*)d_in[19];
  const float* W_m1     = (const float*)d_in[20];
  const float* b_m1     = (const float*)d_in[21];
  const float* W_m2     = (const float*)d_in[22];
  const float* b_m2     = (const float*)d_in[23];
  const float* a_dw_w   = (const float*)d_in[24];
  const float* a_dw_b   = (const float*)d_in[25];
  const float* a_cw     = (const float*)d_in[26];
  const float* a_cb     = (const float*)d_in[27];
  const float* a_g      = (const float*)d_in[28];
  const float* a_b      = (const float*)d_in[29];

  float* w = (float*)d_ws;
  size_t off = 0;
  float* B1   = w + off; off += (size_t)L_ * L_;       // DCT-II matrix (transposed)
  float* B2   = w + off; off += (size_t)L_ * L_;       // DCT-III matrix (transposed)
  float* XT   = w + off; off += (size_t)BD_ * L_;
  float* XRES = w + off; off += (size_t)BD_ * L_;
  float* Z    = w + off; off += (size_t)BD_ * L_;      // z_dct -> masked/gelu -> bn -> zsum
  float* Z1   = w + off; off += (size_t)BD_ * L_;
  float* S    = w + off; off += (size_t)BD_ * L_;      // s -> z2
  float* ZC   = w + off; off += (size_t)BD_ * L_;      // zc_pre -> z2f
  float* WEFF = w + off; off += 512 * PRED_;
  float* BEFF = w + off; off += PRED_;
  float* WS   = w + off; off += N_ * 256;
  float* BS   = w + off; off += N_ * 16;
  float* ZRES = w + off; off += (size_t)BD_ * PRED_;
  float* H    = w + off; off += (size_t)BD_ * PRED_;
  float* ZMLP = w + off; off += (size_t)BD_ * PRED_;
  float* ENER = w + off; off += BD_;
  float* NORM = w + off; off += BD_;
  float* ATT  = w + off; off += BD_;
  float* ATT1 = w + off; off += BD_;
  float* BND  = w + off; off += 2 * D_;
  float* BNP  = w + off; off += 2 * N_;
  float* BNQ  = w + off; off += 2 * N_;
  float* SCAL = w + off; off += 8;

  const int EL = BD_ * L_;                 // 2752512
  const int T = 256;

  // 0) DCT matrices + transpose/residual + folded weights
  k_gen_dct<<<CDIV(L_ * L_, T), T, 0, stream>>>(B1, B2);
  k_transpose<<<CDIV(EL, T), T, 0, stream>>>(x, w_dct, b_dct, XT, XRES);
  k_prep_weff<<<CDIV(512 * PRED_, T), T, 0, stream>>>(W_embed, W_linres, WEFF);
  k_prep_beff<<<1, 128, 0, stream>>>(b_embed, W_linres, b_linres, BEFF);
  k_prep_ws<<<CDIV(N_ * 256, T), T, 0, stream>>>(W_embed, w_dc, W_dres, WS);
  k_prep_bs<<<CDIV(N_ * 16, T), T, 0, stream>>>(b_embed, w_dc, b_dc, W_dres, b_dres, BS);

  // 1) z_dct = xT @ D2^T   (WMMA f32, 5376x512x512)
  launch_gemm(XT, B1, Z, nullptr, nullptr, BD_, L_, L_, 0, stream);
  // 2) adaptive high-freq mask
  k_rowsumsq<<<CDIV(BD_ * 32, T), T, 0, stream>>>(Z, ENER);
  k_median<<<1, B_, 0, stream>>>(ENER, NORM);
  k_quantile<<<1, 1024, 0, stream>>>(NORM, thr_p, SCAL);
  k_mask_gelu<<<CDIV(EL, T), T, 0, stream>>>(Z, NORM, SCAL, w_dct, b_dct);
  // 3) BN over (B,L) per dvar
  k_bn_stats_dct<<<D_, T, 0, stream>>>(Z, BND);
  k_bn_apply_dct<<<CDIV(EL, T), T, 0, stream>>>(Z, BND, g_dctn, b_dctn);
  // 4) z1 = z @ D3^T + x_res   (WMMA f32)
  launch_gemm(Z, B2, Z1, nullptr, XRES, BD_, L_, L_, 0, stream);
  // 5) folded patch path: s = zd + res  (block-diagonal WMMA), then BN/GELU chain
  k_gemm_patch<<<CDIV((BD_ / 16) * N_, 8), T, 0, stream>>>(XT, WS, BS, S);
  k_bn_stats_cols<<<N_, T, 0, stream>>>(S, BNP);
  k_bn_gelu_cols<<<CDIV(EL, T), T, 0, stream>>>(S, BNP, g_patch, b_patch);   // S = z2
  k_dconv<<<CDIV(EL, T), T, 0, stream>>>(S, w_dc1, b_dc1, ZC);
  k_bn_stats_cols<<<N_, T, 0, stream>>>(ZC, BNQ);
  k_bn_gelu_add_cols<<<CDIV(EL, T), T, 0, stream>>>(ZC, S, BNQ, g_depth, b_depth); // ZC = z2f
  // 6) z_res = xT @ W_eff + b_eff   (folded W_embed/W_linres; WMMA f32)
  launch_gemm(XT, WEFF, ZRES, BEFF, nullptr, BD_, L_, PRED_, 0, stream);
  // 7) DctAtt
  k_att<<<CDIV(BD_ * 32, T), T, 0, stream>>>(Z1, ZC, B1, a_dw_w, a_dw_b, ATT);
  k_att_stats<<<1, 1024, 0, stream>>>(ATT, SCAL);
  k_att_softmax<<<1, B_, 0, stream>>>(ATT, SCAL, a_g, a_b, a_cw, a_cb, ATT1);
  k_zsum<<<CDIV(EL, T), T, 0, stream>>>(Z1, ZC, ATT1, Z);                    // Z = zsum
  // 8) MLP: h = gelu(zsum @ W_m1 + b_m1); z_mlp = h @ W_m2 + b_m2  (WMMA f32)
  launch_gemm(Z, W_m1, H, b_m1, nullptr, BD_, L_, PRED_, 1, stream);
  launch_gemm(H, W_m2, ZMLP, b_m2, nullptr, BD_, PRED_, PRED_, 0, stream);
  // 9) out = z_res + z_mlp, transposed to (B, PRED, D)
  k_final<<<CDIV(B_ * PRED_ * D_, T), T, 0, stream>>>(ZRES, ZMLP, (float*)d_out);
}